// GMSDRCell_82703890252033
// MI455X (gfx1250) — compile-verified
//
#include <hip/hip_runtime.h>
#include <hip/hip_bf16.h>

typedef __attribute__((ext_vector_type(2))) float v2f;
typedef __attribute__((ext_vector_type(8))) float v8f;

#if defined(__gfx1250__) && __has_builtin(__builtin_amdgcn_wmma_f32_16x16x4_f32)
#define HAVE_WMMA_F32 1
#endif

// ---------------- constants ----------------
#define BATCH 64
#define KPRE  6
#define NN    2048
#define DD    64
#define FIN   66          // INPUT_DIM + D
#define FB    4224        // FIN * BATCH
#define ND    131072      // NN * DD
#define MATSZ 8650752     // NN * FB

// ---------------- kernel 1: adp = softmax(relu(nv1@nv2), axis=1) ----------------
__global__ __launch_bounds__(256) void k_adp(const float* __restrict__ nv1,
                                             const float* __restrict__ nv2,
                                             float* __restrict__ adp) {
  __shared__ float red[256];
  int row = blockIdx.x, t = threadIdx.x;
  float v1[10];
#pragma unroll
  for (int q = 0; q < 10; ++q) v1[q] = nv1[row * 10 + q];
  float vals[8];
  float mx = -1e30f;
#pragma unroll
  for (int i = 0; i < 8; ++i) {
    int j = t + 256 * i;
    float s = 0.f;
#pragma unroll
    for (int q = 0; q < 10; ++q) s += v1[q] * nv2[q * NN + j];
    s = s > 0.f ? s : 0.f;
    vals[i] = s;
    mx = fmaxf(mx, s);
  }
  red[t] = mx; __syncthreads();
  for (int s2 = 128; s2 > 0; s2 >>= 1) { if (t < s2) red[t] = fmaxf(red[t], red[t + s2]); __syncthreads(); }
  mx = red[0]; __syncthreads();
  float sum = 0.f;
#pragma unroll
  for (int i = 0; i < 8; ++i) { vals[i] = __expf(vals[i] - mx); sum += vals[i]; }
  red[t] = sum; __syncthreads();
  for (int s2 = 128; s2 > 0; s2 >>= 1) { if (t < s2) red[t] += red[t + s2]; __syncthreads(); }
  float inv = 1.f / red[0];
#pragma unroll
  for (int i = 0; i < 8; ++i) adp[(long)row * NN + t + 256 * i] = vals[i] * inv;
}

// ---------------- kernel 2: build x0 (n, f*B + b) ----------------
__global__ __launch_bounds__(256) void k_build_x0(const float* __restrict__ inputs,
                                                  const float* __restrict__ hx,
                                                  float* __restrict__ x0) {
  long idx = (long)blockIdx.x * 256 + threadIdx.x;  // n*FB + f*64 + b, total NN*FIN*BATCH
  int b = (int)(idx & 63);
  int f = (int)((idx >> 6) % FIN);
  int n = (int)(idx / (FIN * BATCH));
  float v;
  if (f < 2) v = inputs[(long)b * (NN * 2) + n * 2 + f];
  else       v = hx[(((long)b * KPRE + (KPRE - 1)) * NN + n) * DD + (f - 2)];
  x0[idx] = v;
}

// ---------------- kernel 3: fp32 WMMA GEMM  Y = S @ X  (+ epilogue) ----------------
// mode 0: Y = acc ; mode 1: Y = 2*acc - Z ; mode 2: Y = acc - Z
// Block tile 256x64, 8 waves; each wave owns TWO 16-row tiles x four 16-col tiles,
// so every B fragment is reused by two WMMAs (0.75 LDS loads per WMMA).
#define BM 256
#define BCOL 64
#define BK 32
#define LDP (BK + 4)   // 36 floats: 16B-aligned rows, conflict-skewed

__global__ __launch_bounds__(256) void k_gemm(const float* __restrict__ S,
                                              const float* __restrict__ X,
                                              const float* __restrict__ Z,
                                              float* __restrict__ Y,
                                              int mode) {
  __shared__ float sS[BM][LDP];     // S tile, row-major
  __shared__ float sX[BCOL][LDP];   // X tile, TRANSPOSED (col-major) -> b64 fragment loads
  const int t = threadIdx.x;
  const int wave = t >> 5, lane = t & 31;
  const int m_blk = blockIdx.y * BM;
  const int c_blk = blockIdx.x * BCOL;

  v8f acc[2][4];
#pragma unroll
  for (int r = 0; r < 2; ++r)
#pragma unroll
    for (int c = 0; c < 4; ++c) acc[r][c] = v8f{};

  const int mrow0 = wave * 32 + (lane & 15);   // first 16-row tile
  const int mrow1 = mrow0 + 16;                // second 16-row tile
  const int khalf = (lane >> 4) * 2;   // lanes 0-15 -> K {0,1}, lanes 16-31 -> K {2,3}
  const int ncol  = lane & 15;
  const int halfsel = lane >> 4;

  for (int kc = 0; kc < NN; kc += BK) {
    // S tile: 256x32 floats = 2048 float4, 8 per thread (coalesced rows)
#pragma unroll
    for (int i = 0; i < 8; ++i) {
      int lin = t + 256 * i;               // float4 units
      int r = lin >> 3, kk = (lin & 7) << 2;
      float4 v = *(const float4*)(S + (long)(m_blk + r) * NN + kc + kk);
      *(float4*)(&sS[r][kk]) = v;
    }
    // X tile: 32x64 floats = 512 float4, 2 per thread; store transposed
#pragma unroll
    for (int i = 0; i < 2; ++i) {
      int lin = t + 256 * i;               // float4 units
      int kk = lin >> 4, c4 = (lin & 15) << 2;
      float4 v = *(const float4*)(X + (long)(kc + kk) * FB + c_blk + c4);
      sX[c4 + 0][kk] = v.x; sX[c4 + 1][kk] = v.y;
      sX[c4 + 2][kk] = v.z; sX[c4 + 3][kk] = v.w;
    }
    __syncthreads();

#pragma unroll
    for (int k0 = 0; k0 < BK; k0 += 4) {
#if defined(HAVE_WMMA_F32)
      v2f a0 = *(const v2f*)(&sS[mrow0][k0 + khalf]);
      v2f a1 = *(const v2f*)(&sS[mrow1][k0 + khalf]);
#pragma unroll
      for (int c = 0; c < 4; ++c) {
        v2f bb = *(const v2f*)(&sX[c * 16 + ncol][k0 + khalf]);
        acc[0][c] = __builtin_amdgcn_wmma_f32_16x16x4_f32(
            false, a0, false, bb, (short)0, acc[0][c], false, false);
        acc[1][c] = __builtin_amdgcn_wmma_f32_16x16x4_f32(
            false, a1, false, bb, (short)0, acc[1][c], false, false);
      }
#else
      // layout-identical VALU fallback: lane owns col (c*16+ncol), rows v + halfsel*8
#pragma unroll
      for (int kk = 0; kk < 4; ++kk) {
#pragma unroll
        for (int c = 0; c < 4; ++c) {
          float bv = sX[c * 16 + ncol][k0 + kk];
#pragma unroll
          for (int v = 0; v < 8; ++v) {
            acc[0][c][v] += sS[wave * 32 + v + halfsel * 8][k0 + kk] * bv;
            acc[1][c][v] += sS[wave * 32 + 16 + v + halfsel * 8][k0 + kk] * bv;
          }
        }
      }
#endif
    }
    __syncthreads();
  }

  // epilogue: C layout -> VGPR v, lanes 0-15: M=v ; lanes 16-31: M=v+8 ; N = lane&15
#pragma unroll
  for (int rt = 0; rt < 2; ++rt) {
#pragma unroll
    for (int c = 0; c < 4; ++c) {
#pragma unroll
      for (int v = 0; v < 8; ++v) {
        int row = m_blk + wave * 32 + rt * 16 + v + halfsel * 8;
        int col = c_blk + c * 16 + ncol;
        long o = (long)row * FB + col;
        float y = acc[rt][c][v];
        if (mode == 1)      y = 2.f * y - Z[o];
        else if (mode == 2) y = y - Z[o];
        Y[o] = y;
      }
    }
  }
}

// ---------------- kernel 4: gconv (x@gconv_w + gb, leaky) then conv@W + bias ----------------
__global__ __launch_bounds__(256) void k_gconv(const float* __restrict__ m0, const float* __restrict__ m1,
                                               const float* __restrict__ m2, const float* __restrict__ m3,
                                               const float* __restrict__ m4,
                                               const float* __restrict__ gw, const float* __restrict__ gb,
                                               const float* __restrict__ W,  const float* __restrict__ biasN,
                                               float* __restrict__ out1) {
  __shared__ float xs_l[55 * 64];     // 11 f's * 5 mats * 64 b
  __shared__ float gw_l[55 * 64];
  __shared__ float conv_l[64 * 64];
  __shared__ float w_l[64 * 64];
  const int node = blockIdx.x, t = threadIdx.x;
  const int bb = t >> 2;          // batch 0..63
  const int d0 = t & 3;
  float acc[16];
#pragma unroll
  for (int j = 0; j < 16; ++j) acc[j] = 0.f;

  for (int ch = 0; ch < 6; ++ch) {
    int f0 = ch * 11;
#pragma unroll
    for (int m = 0; m < 5; ++m) {
      const float* mp = (m == 0) ? m0 : (m == 1) ? m1 : (m == 2) ? m2 : (m == 3) ? m3 : m4;
      for (int i = t; i < 11 * 64; i += 256) {
        int b_ = i & 63, f = i >> 6;
        xs_l[(f * 5 + m) * 64 + b_] = mp[(long)node * FB + (f0 + f) * 64 + b_];
      }
    }
    for (int i = t; i < 55 * 64; i += 256) gw_l[i] = gw[(long)f0 * 5 * 64 + i];
    __syncthreads();
    for (int r = 0; r < 55; ++r) {
      float xv = xs_l[r * 64 + bb];
#pragma unroll
      for (int j = 0; j < 16; ++j) acc[j] += xv * gw_l[r * 64 + d0 + 4 * j];
    }
    __syncthreads();
  }
#pragma unroll
  for (int j = 0; j < 16; ++j) {
    int d = d0 + 4 * j;
    float c = acc[j] + gb[d];
    conv_l[bb * 64 + d] = (c >= 0.f) ? c : 0.01f * c;
  }
  for (int i = t; i < 4096; i += 256) w_l[i] = W[i];
  __syncthreads();
#pragma unroll 1
  for (int j = 0; j < 16; ++j) {
    int d = d0 + 4 * j;
    float o = biasN[(long)node * 64 + d];
    for (int dp = 0; dp < 64; ++dp) o += conv_l[bb * 64 + dp] * w_l[dp * 64 + d];
    out1[((long)bb * NN + node) * DD + d] = o;
  }
}

// ---------------- kernel 5: attention scores ----------------
__global__ __launch_bounds__(256) void k_scores(const float* __restrict__ hx, const float* __restrict__ R,
                                                const float* __restrict__ aw, const float* __restrict__ ab,
                                                float* __restrict__ scores) {
  __shared__ float red[256];
  int bk = blockIdx.x;          // b*6 + k
  int k_ = bk % KPRE;
  const float* h = hx + (long)bk * ND;
  const float* r = R + (long)k_ * ND;
  float s = 0.f;
  for (int i = threadIdx.x; i < ND; i += 256) s += (h[i] + r[i]) * aw[i];
  red[threadIdx.x] = s; __syncthreads();
  for (int st = 128; st > 0; st >>= 1) { if (threadIdx.x < st) red[threadIdx.x] += red[threadIdx.x + st]; __syncthreads(); }
  if (threadIdx.x == 0) scores[bk] = red[0] + ab[0];
}

// ---------------- kernel 6: softmax(k) + attention + output + hx_k_new ----------------
__global__ __launch_bounds__(256) void k_final(const float* __restrict__ hx, const float* __restrict__ R,
                                               const float* __restrict__ scores, const float* __restrict__ out1,
                                               float* __restrict__ dout) {
  long idx = (long)blockIdx.x * 256 + threadIdx.x;   // < BATCH*ND
  long b_ = idx >> 17;
  long i2 = idx & (ND - 1);
  float sc[6], w[6], mx = -1e30f;
#pragma unroll
  for (int k = 0; k < KPRE; ++k) { sc[k] = scores[b_ * KPRE + k]; mx = fmaxf(mx, sc[k]); }
  float sum = 0.f;
#pragma unroll
  for (int k = 0; k < KPRE; ++k) { w[k] = __expf(sc[k] - mx); sum += w[k]; }
  float inv = 1.f / sum;
  float h[6], att = 0.f;
#pragma unroll
  for (int k = 0; k < KPRE; ++k) {
    h[k] = hx[(b_ * KPRE + k) * ND + i2];
    att += (h[k] + R[(long)k * ND + i2]) * (w[k] * inv);
  }
  float outp = out1[idx] + att;
  dout[idx] = outp;
  float* hxn = dout + (long)BATCH * ND;
#pragma unroll
  for (int k = 0; k < KPRE - 1; ++k) hxn[(b_ * KPRE + k) * ND + i2] = h[k + 1];
  hxn[(b_ * KPRE + (KPRE - 1)) * ND + i2] = outp;
}

// ---------------- launcher ----------------
extern "C" void kernel_launch(void* const* d_in, const int* in_sizes, int n_in,
                              void* d_out, int out_size, void* d_ws, size_t ws_size,
                              hipStream_t stream) {
  const float* inputs  = (const float*)d_in[0];
  const float* hx_k    = (const float*)d_in[1];
  const float* support = (const float*)d_in[2];
  const float* nv1     = (const float*)d_in[3];
  const float* nv2     = (const float*)d_in[4];
  const float* biasN   = (const float*)d_in[5];
  const float* R       = (const float*)d_in[6];
  const float* att_w   = (const float*)d_in[7];
  const float* att_b   = (const float*)d_in[8];
  const float* gconv_w = (const float*)d_in[9];
  const float* gconv_b = (const float*)d_in[10];
  const float* W       = (const float*)d_in[11];

  float* ws   = (float*)d_ws;
  float* adp  = ws;                              // 4,194,304
  float* x0   = adp + (long)NN * NN;             // 8,650,752 each
  float* x1   = x0 + MATSZ;
  float* x2   = x1 + MATSZ;
  float* x1a  = x2 + MATSZ;
  float* x2a  = x1a + MATSZ;
  float* out1 = x2a + MATSZ;                     // 8,388,608
  float* scores = out1 + (long)BATCH * ND;       // 384
  float* dout = (float*)d_out;

  k_adp<<<NN, 256, 0, stream>>>(nv1, nv2, adp);
  k_build_x0<<<(NN * FIN * BATCH) / 256, 256, 0, stream>>>(inputs, hx_k, x0);

  dim3 gg(FB / BCOL, NN / BM);   // (66, 8)
  k_gemm<<<gg, 256, 0, stream>>>(support, x0, x0, x1, 0);   // x1  = S @ x0
  k_gemm<<<gg, 256, 0, stream>>>(support, x1, x0, x2, 1);   // x2  = 2*(S@x1) - x0
  k_gemm<<<gg, 256, 0, stream>>>(adp,     x1, x0, x1a, 0);  // x1a = adp @ x1
  k_gemm<<<gg, 256, 0, stream>>>(adp,    x1a, x1, x2a, 2);  // x2a = adp @ x1a - x1

  k_gconv<<<NN, 256, 0, stream>>>(x0, x1, x2, x1a, x2a, gconv_w, gconv_b, W, biasN, out1);
  k_scores<<<BATCH * KPRE, 256, 0, stream>>>(hx_k, R, att_w, att_b, scores);
  k_final<<<(BATCH * ND) / 256, 256, 0, stream>>>(hx_k, R, scores, out1, dout);
}